// CoordinatesToSpikes_84731114815776
// MI455X (gfx1250) — compile-verified
//
#include <hip/hip_runtime.h>

// CoordinatesToSpikes: expand (B=256, C=256) coordinates into a one-hot
// spike train (B, T=1000, C). Pure streaming-store problem:
//   262 MB out / 23.3 TB/s  ->  ~11.3 us floor on MI455X.
// Single pass: each float4 is materialized as (idx == t) ? 1 : 0 and
// streamed with non-temporal B128 stores (output > 192 MB L2, never re-read).
// No matrix math exists in this op, so WMMA is intentionally not used.

typedef __attribute__((ext_vector_type(4))) float v4f;

namespace {
constexpr int   kB = 256;              // batch
constexpr int   kC = 256;              // channels
constexpr int   kT = 1000;             // SEQ_LENGTH
constexpr int   kTPerBlock = 40;       // 25 * 40 == 1000
constexpr int   kRowsPerThread = 10;   // ty strip: 4 * 10 == 40
// Reference math (float32): times = T_EARLY + v*(T_LATE-T_EARLY); idx = rint(times/DT)
constexpr float kTEarly = 2e-06f;
constexpr float kScale  = (float)(0.0008 - 2e-06);   // 0.000798
constexpr float kInvDt  = 1.0f / 1e-06f;             // == 1e6f exactly (15625 * 2^6)
}

__global__ __launch_bounds__(256)
void spikes_onehot_kernel(const float* __restrict__ coord,
                          float* __restrict__ out) {
  const int b  = blockIdx.x;           // batch index
  const int tx = threadIdx.x;          // 0..63 : 4 consecutive channels each
  const int ty = threadIdx.y;          // 0..3  : 10-row strip each (wave-uniform)
  const int c0 = tx << 2;

  // Contiguous 10-row strip per ty slice -> compile-time trip count -> full
  // unroll into 10 global_store_b128 with immediate offsets off one base VGPR.
  const int tBase = blockIdx.y * kTPerBlock + ty * kRowsPerThread;
  float* outRow = out + ((size_t)b * kT + tBase) * kC + c0;

  // idx = rint(v*798 + 2) is provably in [2, 800] for v in [0,1):
  // strips entirely above 800 are pure zero-fill (t = 801..999, ~20% of output).
  if (tBase > 800) {
    const v4f z = {0.0f, 0.0f, 0.0f, 0.0f};
#pragma unroll
    for (int k = 0; k < kRowsPerThread; ++k)
      __builtin_nontemporal_store(z, reinterpret_cast<v4f*>(outRow + (size_t)k * kC));
    return;
  }

  // One B128 load of this lane's 4 coordinate values (tiny, L2-resident).
  const v4f v = *reinterpret_cast<const v4f*>(coord + (size_t)b * kC + c0);

  // Spike time indices (v_rndne_f32 = round-half-to-even == jnp.round).
  // Divide-by-DT replaced with multiply by the exact f32 reciprocal (1e6f):
  // avoids 4x full IEEE v_div_scale/div_fmas sequences seen in round 1.
  const int i0 = (int)rintf((kTEarly + v.x * kScale) * kInvDt);
  const int i1 = (int)rintf((kTEarly + v.y * kScale) * kInvDt);
  const int i2 = (int)rintf((kTEarly + v.z * kScale) * kInvDt);
  const int i3 = (int)rintf((kTEarly + v.w * kScale) * kInvDt);

#pragma unroll
  for (int k = 0; k < kRowsPerThread; ++k) {
    const int t = tBase + k;
    v4f r;
    r.x = (i0 == t) ? 1.0f : 0.0f;
    r.y = (i1 == t) ? 1.0f : 0.0f;
    r.z = (i2 == t) ? 1.0f : 0.0f;
    r.w = (i3 == t) ? 1.0f : 0.0f;
    // Non-temporal streaming store: global_store_b128 ... th:TH_STORE_NT
    __builtin_nontemporal_store(r, reinterpret_cast<v4f*>(outRow + (size_t)k * kC));
  }
}

extern "C" void kernel_launch(void* const* d_in, const int* in_sizes, int n_in,
                              void* d_out, int out_size, void* d_ws, size_t ws_size,
                              hipStream_t stream) {
  (void)in_sizes; (void)n_in; (void)d_ws; (void)ws_size; (void)out_size;
  const float* coord = (const float*)d_in[0];   // (256, 256) float32
  float* out = (float*)d_out;                   // (256, 1000, 256) float32

  dim3 grid(kB, kT / kTPerBlock);  // 256 x 25 = 6400 blocks
  dim3 block(64, 4);               // 256 threads = 8 wave32s; 64 lanes * f4 = 256 ch
  spikes_onehot_kernel<<<grid, block, 0, stream>>>(coord, out);
}